// GCN_17136919511827
// MI455X (gfx1250) — compile-verified
//
#include <hip/hip_runtime.h>
#include <hip/hip_bf16.h>
#include <math.h>

typedef __attribute__((ext_vector_type(2))) float v2f;
typedef __attribute__((ext_vector_type(8))) float v8f;

#define F_IN 256
#define HID  128
#define NCLS 40

// ---------------------------------------------------------------------------
// Degree / normalization kernels
// ---------------------------------------------------------------------------
__global__ void init_deg_k(float* __restrict__ deg, int n) {
    int i = blockIdx.x * blockDim.x + threadIdx.x;
    if (i < n) deg[i] = 1.0f;  // self-loop contributes 1
}

__global__ void deg_accum_k(const long long* __restrict__ dst,
                            float* __restrict__ deg, long long E) {
    long long i = (long long)blockIdx.x * blockDim.x + threadIdx.x;
    long long stride = (long long)gridDim.x * blockDim.x;
    for (long long e = i; e < E; e += stride)
        atomicAdd(&deg[dst[e]], 1.0f);
}

__global__ void rsqrt_k(float* __restrict__ deg, int n) {
    int i = blockIdx.x * blockDim.x + threadIdx.x;
    if (i < n) {
        float d = deg[i];
        deg[i] = (d > 0.0f) ? rsqrtf(d) : 0.0f;  // deg -> dinv in place
    }
}

// ---------------------------------------------------------------------------
// GEMM1: xw[N,128] = x[N,256] @ W1[256,128], fp32 WMMA 16x16x4
// 1 block = one 16-row tile; 8 waves each own a 16-col slab (8*16 = 128 = HID)
// ---------------------------------------------------------------------------
__global__ __launch_bounds__(256) void gemm1_wmma_k(
    const float* __restrict__ x, const float* __restrict__ W1,
    float* __restrict__ xw) {
    __shared__ float As[16][F_IN + 4];  // 16x260 f32, row stride 1040B (16B aligned)

    const int m0 = blockIdx.x * 16;
    // Cooperatively stage A tile (16x256) as float4: 1024 vec4 / 256 threads
    {
        const int t = threadIdx.x;
#pragma unroll
        for (int i = 0; i < 4; ++i) {
            int q   = t + i * 256;      // 0..1023
            int row = q >> 6;           // 64 float4 per row
            int c4  = q & 63;
            float4 v = *(const float4*)(x + (size_t)(m0 + row) * F_IN + c4 * 4);
            *(float4*)(&As[row][c4 * 4]) = v;
        }
    }
    __syncthreads();

    const int lane = threadIdx.x & 31;
    const int wv   = threadIdx.x >> 5;   // 0..7
    const int n0   = wv * 16;
    const int l    = lane & 15;
    const int half = lane >> 4;          // 0: K 0..1, 1: K 2..3

    v8f c = {};
    for (int k = 0; k < F_IN; k += 4) {
        v2f a, b;
        // A 16x4 f32 layout: lanes 0-15 -> K=k,k+1 ; lanes 16-31 -> K=k+2,k+3
        a.x = As[l][k + 2 * half + 0];
        a.y = As[l][k + 2 * half + 1];
        // B 4x16 f32 layout (mirror): row K, col = n0 + l
        b.x = W1[(size_t)(k + 2 * half + 0) * HID + n0 + l];
        b.y = W1[(size_t)(k + 2 * half + 1) * HID + n0 + l];
        c = __builtin_amdgcn_wmma_f32_16x16x4_f32(false, a, false, b,
                                                  (short)0, c, false, false);
    }
    // C/D layout: VGPR r -> row (r + 8*half), col = lane%16
#pragma unroll
    for (int r = 0; r < 8; ++r)
        xw[(size_t)(m0 + r + 8 * half) * HID + n0 + l] = c[r];
}

// ---------------------------------------------------------------------------
// GEMM2: hw[N,40] = hid[N,128] @ W2[128,40], fp32 WMMA 16x16x4
// 1 block = 128 threads; waves 0..2 own col tiles n0 = 0,16,32 (guard at 40)
// ---------------------------------------------------------------------------
__global__ __launch_bounds__(128) void gemm2_wmma_k(
    const float* __restrict__ hid, const float* __restrict__ W2,
    float* __restrict__ hw) {
    const int wv = threadIdx.x >> 5;
    if (wv >= 3) return;  // no barriers below; safe early-exit

    const int m0   = blockIdx.x * 16;
    const int lane = threadIdx.x & 31;
    const int n0   = wv * 16;
    const int l    = lane & 15;
    const int half = lane >> 4;
    const int col  = n0 + l;
    const bool cv  = (col < NCLS);

    v8f c = {};
    for (int k = 0; k < HID; k += 4) {
        v2f a, b;
        a.x = hid[(size_t)(m0 + l) * HID + k + 2 * half + 0];
        a.y = hid[(size_t)(m0 + l) * HID + k + 2 * half + 1];
        b.x = cv ? W2[(size_t)(k + 2 * half + 0) * NCLS + col] : 0.0f;
        b.y = cv ? W2[(size_t)(k + 2 * half + 1) * NCLS + col] : 0.0f;
        c = __builtin_amdgcn_wmma_f32_16x16x4_f32(false, a, false, b,
                                                  (short)0, c, false, false);
    }
    if (cv) {
#pragma unroll
        for (int r = 0; r < 8; ++r)
            hw[(size_t)(m0 + r + 8 * half) * NCLS + col] = c[r];
    }
}

// ---------------------------------------------------------------------------
// Aggregation: self-loop init, edge scatter-add (wave per edge), bias/act
// ---------------------------------------------------------------------------
__global__ void selfinit1_k(const float* __restrict__ xw,
                            const float* __restrict__ dinv,
                            float* __restrict__ agg, int total) {
    int i = blockIdx.x * blockDim.x + threadIdx.x;
    if (i < total) {
        float di = dinv[i >> 7];  // node = i / 128
        agg[i] = xw[i] * di * di;
    }
}

__global__ void scatter1_k(const long long* __restrict__ src,
                           const long long* __restrict__ dst,
                           const float* __restrict__ dinv,
                           const float* __restrict__ xw,
                           float* __restrict__ agg, long long E) {
    long long wid = (long long)blockIdx.x * (blockDim.x >> 5) + (threadIdx.x >> 5);
    long long nw  = (long long)gridDim.x * (blockDim.x >> 5);
    int lane = threadIdx.x & 31;
    for (long long e = wid; e < E; e += nw) {
        long long s = src[e], d = dst[e];
        float nrm = dinv[s] * dinv[d];
        float4 v = ((const float4*)(xw + (size_t)s * HID))[lane];  // 32 lanes x 4 = 128
        float* o = agg + (size_t)d * HID + lane * 4;
        atomicAdd(o + 0, v.x * nrm);
        atomicAdd(o + 1, v.y * nrm);
        atomicAdd(o + 2, v.z * nrm);
        atomicAdd(o + 3, v.w * nrm);
    }
}

__global__ void bias_relu_k(const float* __restrict__ agg,
                            const float* __restrict__ b,
                            float* __restrict__ out, int total) {
    int i = blockIdx.x * blockDim.x + threadIdx.x;
    if (i < total) out[i] = fmaxf(agg[i] + b[i & (HID - 1)], 0.0f);
}

__global__ void selfinit2_k(const float* __restrict__ hw,
                            const float* __restrict__ dinv,
                            float* __restrict__ agg, int total) {
    int i = blockIdx.x * blockDim.x + threadIdx.x;
    if (i < total) {
        float di = dinv[i / NCLS];
        agg[i] = hw[i] * di * di;
    }
}

__global__ void scatter2_k(const long long* __restrict__ src,
                           const long long* __restrict__ dst,
                           const float* __restrict__ dinv,
                           const float* __restrict__ hw,
                           float* __restrict__ agg, long long E) {
    long long wid = (long long)blockIdx.x * (blockDim.x >> 5) + (threadIdx.x >> 5);
    long long nw  = (long long)gridDim.x * (blockDim.x >> 5);
    int lane = threadIdx.x & 31;
    for (long long e = wid; e < E; e += nw) {
        long long s = src[e], d = dst[e];
        float nrm = dinv[s] * dinv[d];
        const float* hs = hw + (size_t)s * NCLS;
        float* o = agg + (size_t)d * NCLS;
        for (int f = lane; f < NCLS; f += 32)
            atomicAdd(o + f, hs[f] * nrm);
    }
}

__global__ void bias2_k(const float* __restrict__ agg,
                        const float* __restrict__ b,
                        float* __restrict__ out, int total) {
    int i = blockIdx.x * blockDim.x + threadIdx.x;
    if (i < total) out[i] = agg[i] + b[i % NCLS];
}

// ---------------------------------------------------------------------------
// log_softmax over 40 classes, one wave per row
// ---------------------------------------------------------------------------
__global__ void logsoftmax_k(const float* __restrict__ x2,
                             float* __restrict__ logp, int N) {
    int wid  = (blockIdx.x * blockDim.x + threadIdx.x) >> 5;
    int lane = threadIdx.x & 31;
    if (wid >= N) return;
    const float* r = x2 + (size_t)wid * NCLS;
    float v0 = r[lane];                                    // lanes 0..31
    float v1 = (lane < 8) ? r[32 + lane] : -INFINITY;      // lanes 0..7 -> 32..39
    float m = fmaxf(v0, v1);
    for (int off = 16; off; off >>= 1) m = fmaxf(m, __shfl_xor(m, off, 32));
    float s = expf(v0 - m) + ((lane < 8) ? expf(v1 - m) : 0.0f);
    for (int off = 16; off; off >>= 1) s += __shfl_xor(s, off, 32);
    float lse = m + logf(s);
    float* o = logp + (size_t)wid * NCLS;
    o[lane] = v0 - lse;
    if (lane < 8) o[32 + lane] = v1 - lse;
}

// ---------------------------------------------------------------------------
extern "C" void kernel_launch(void* const* d_in, const int* in_sizes, int n_in,
                              void* d_out, int out_size, void* d_ws, size_t ws_size,
                              hipStream_t stream) {
    const float*     x   = (const float*)d_in[0];
    const long long* ei  = (const long long*)d_in[1];  // int64 [2, E]
    const float*     W1  = (const float*)d_in[2];
    const float*     b1  = (const float*)d_in[3];
    const float*     W2  = (const float*)d_in[4];
    const float*     b2  = (const float*)d_in[5];

    const int       N = in_sizes[0] / F_IN;          // 50000 (divisible by 16)
    const long long E = (long long)in_sizes[1] / 2;  // 1.6M
    const long long* srcIdx = ei;
    const long long* dstIdx = ei + E;

    // Workspace layout (floats)
    float* ws = (float*)d_ws;
    size_t o = 0;
    float* deg  = ws + o; o += ((size_t)N + 63) & ~(size_t)63;  // deg -> dinv in place
    float* xw   = ws + o; o += (size_t)N * HID;
    float* agg1 = ws + o; o += (size_t)N * HID;
    float* hw   = ws + o; o += (size_t)N * NCLS;
    float* agg2 = ws + o; o += (size_t)N * NCLS;

    // Output layout: hid [N,128] | x2 [N,40] | logp [N,40]
    float* outp = (float*)d_out;
    float* hid  = outp;
    float* x2   = outp + (size_t)N * HID;
    float* logp = x2   + (size_t)N * NCLS;

    const int tHID = N * HID;   // 6.4M
    const int tCLS = N * NCLS;  // 2.0M

    // 1) degrees -> dinv
    init_deg_k<<<(N + 255) / 256, 256, 0, stream>>>(deg, N);
    deg_accum_k<<<(unsigned)((E + 255) / 256), 256, 0, stream>>>(dstIdx, deg, E);
    rsqrt_k<<<(N + 255) / 256, 256, 0, stream>>>(deg, N);

    // 2) layer 1
    gemm1_wmma_k<<<N / 16, 256, 0, stream>>>(x, W1, xw);
    selfinit1_k<<<(tHID + 255) / 256, 256, 0, stream>>>(xw, deg, agg1, tHID);
    scatter1_k<<<(unsigned)((E + 7) / 8), 256, 0, stream>>>(srcIdx, dstIdx, deg, xw, agg1, E);
    bias_relu_k<<<(tHID + 255) / 256, 256, 0, stream>>>(agg1, b1, hid, tHID);

    // 3) layer 2
    gemm2_wmma_k<<<N / 16, 128, 0, stream>>>(hid, W2, hw);
    selfinit2_k<<<(tCLS + 255) / 256, 256, 0, stream>>>(hw, deg, agg2, tCLS);
    scatter2_k<<<(unsigned)((E + 7) / 8), 256, 0, stream>>>(srcIdx, dstIdx, deg, hw, agg2, E);
    bias2_k<<<(tCLS + 255) / 256, 256, 0, stream>>>(agg2, b2, x2, tCLS);

    // 4) log-softmax
    logsoftmax_k<<<(N + 7) / 8, 256, 0, stream>>>(x2, logp, N);
}